// GraphSAGEModel_77498389889594
// MI455X (gfx1250) — compile-verified
//
#include <hip/hip_runtime.h>

// GraphSAGE (2-layer) + linear classifier for gfx1250.
// Strategy: edge scatter-mean via f32 atomics (bandwidth bound, ~1.6GB total),
// node GEMMs via full-precision V_WMMA_F32_16X16X4_F32 (GEMM flops are trivial,
// so fp32 WMMA keeps accuracy for free). Classifier fused into layer-2 GEMM.

typedef __attribute__((ext_vector_type(2))) float v2f;
typedef __attribute__((ext_vector_type(8))) float v8f;

__device__ __forceinline__ v8f wmma4(v2f a, v2f b, v8f c) {
    // D = A(16x4 f32) * B(4x16 f32) + C(16x16 f32)
    return __builtin_amdgcn_wmma_f32_16x16x4_f32(
        /*neg_a=*/false, a, /*neg_b=*/false, b,
        /*c_mod=*/(short)0, c, /*reuse_a=*/false, /*reuse_b=*/false);
}

// ---------------------------------------------------------------------------
// Edge scatter: agg[dst] += feat[src]; deg[dst] += 1 (optional).
// 16 threads per edge, 4 channels each (float4 gather, 4x f32 atomic scatter).
// ---------------------------------------------------------------------------
__global__ __launch_bounds__(256)
void scatter_accum(const float* __restrict__ feat,
                   const int* __restrict__ src,
                   const int* __restrict__ dst,
                   float* __restrict__ agg,
                   float* __restrict__ deg,
                   int n_edges) {
    const int tid = blockIdx.x * blockDim.x + threadIdx.x;
    const int e   = tid >> 4;
    if (e >= n_edges) return;
    const int c4 = (tid & 15) << 2;
    const int s = src[e];
    const int d = dst[e];
    const float4 v = *(const float4*)(feat + (size_t)s * 64 + c4);
    float* o = agg + (size_t)d * 64 + c4;
    atomicAdd(o + 0, v.x);
    atomicAdd(o + 1, v.y);
    atomicAdd(o + 2, v.z);
    atomicAdd(o + 3, v.w);
    if (deg != nullptr && c4 == 0) atomicAdd(deg + d, 1.0f);
}

// ---------------------------------------------------------------------------
// out[i] = bc[0]
// ---------------------------------------------------------------------------
__global__ void init_out(float* __restrict__ out, const float* __restrict__ bc, int n) {
    const int i = blockIdx.x * blockDim.x + threadIdx.x;
    if (i < n) out[i] = bc[0];
}

// ---------------------------------------------------------------------------
// h = relu( (agg/max(deg,1)) @ Wl + xroot @ Wr + b )        [N,64]
// 8 waves/block: wave computes a 16(M)x16(N) tile; waves 0-3 / 4-7 handle the
// two M-tiles of each block iteration; n0 = (wave&3)*16.
// Weights staged global -> LDS -> registers once per block.
// ---------------------------------------------------------------------------
__global__ __launch_bounds__(256)
void sage_gemm(const float* __restrict__ xroot,
               const float* __restrict__ agg,
               const float* __restrict__ deg,
               const float* __restrict__ Wl,
               const float* __restrict__ Wr,
               const float* __restrict__ bias,
               float* __restrict__ hout,
               int n_nodes, int n_tiles) {
    __shared__ float lWl[64 * 64];
    __shared__ float lWr[64 * 64];
    __shared__ float lb[64];

    const int tid = threadIdx.x;
    {   // cooperative weight stage (16 KB each)
        const float4* gl = (const float4*)Wl;
        const float4* gr = (const float4*)Wr;
        float4* sl = (float4*)lWl;
        float4* sr = (float4*)lWr;
        for (int i = tid; i < 1024; i += 256) { sl[i] = gl[i]; sr[i] = gr[i]; }
        if (tid < 64) lb[tid] = bias[tid];
    }
    __syncthreads();

    const int lane = tid & 31;
    const int wave = tid >> 5;
    const int n0   = (wave & 3) * 16;
    const int mg   = wave >> 2;          // 0 or 1: which M-tile of the pair
    const int col  = lane & 15;
    const int kp   = (lane >> 4) * 2;    // K sub-pair per ISA A/B layout

    // B slices in registers: 64 VGPRs (bl0/bl1/br0/br1 x16 k-steps)
    float bl0[16], bl1[16], br0[16], br1[16];
#pragma unroll
    for (int ks = 0; ks < 16; ++ks) {
        const int k = 4 * ks + kp;
        bl0[ks] = lWl[(k + 0) * 64 + n0 + col];
        bl1[ks] = lWl[(k + 1) * 64 + n0 + col];
        br0[ks] = lWr[(k + 0) * 64 + n0 + col];
        br1[ks] = lWr[(k + 1) * 64 + n0 + col];
    }
    const float bv = lb[n0 + col];

    for (int tile = blockIdx.x * 2 + mg; tile < n_tiles; tile += gridDim.x * 2) {
        const int m0 = tile * 16;
        int rowi = m0 + col;
        if (rowi >= n_nodes) rowi = n_nodes - 1;
        const float inv = 1.0f / fmaxf(deg[rowi], 1.0f);
        const float* ap = agg   + (size_t)rowi * 64 + kp;
        const float* xp = xroot + (size_t)rowi * 64 + kp;

        const int nt = tile + gridDim.x * 2;
        if (nt < n_tiles) {
            __builtin_prefetch(agg   + (size_t)(nt * 16 + col) * 64, 0, 1);
            __builtin_prefetch(xroot + (size_t)(nt * 16 + col) * 64, 0, 1);
        }

        v8f acc = {};
#pragma unroll
        for (int ks = 0; ks < 16; ++ks) {
            v2f a = *(const v2f*)(ap + 4 * ks);   // A[row][4ks+kp .. +1]
            a *= inv;                             // fold mean normalization
            v2f bl = { bl0[ks], bl1[ks] };
            acc = wmma4(a, bl, acc);
            v2f ax = *(const v2f*)(xp + 4 * ks);
            v2f br = { br0[ks], br1[ks] };
            acc = wmma4(ax, br, acc);
        }

        const int mhi = (lane >> 4) * 8;          // C layout: M = r + 8*(lane>=16)
#pragma unroll
        for (int r = 0; r < 8; ++r) {
            const int row = m0 + mhi + r;
            if (row < n_nodes) {
                const float v = fmaxf(acc[r] + bv, 0.0f);
                hout[(size_t)row * 64 + n0 + col] = v;
            }
        }
    }
}

// ---------------------------------------------------------------------------
// Layer-2 GEMM with fused classifier:
//   h2 = relu(aggN @ Wl + h1 @ Wr + b);  out[row] += sum_n h2[row][n]*Wc[n]
// Cross-lane shfl_xor reduction over the 16-lane N-slice, then one atomicAdd.
// ---------------------------------------------------------------------------
__global__ __launch_bounds__(256)
void sage_gemm_cls(const float* __restrict__ xroot,
                   const float* __restrict__ agg,
                   const float* __restrict__ deg,
                   const float* __restrict__ Wl,
                   const float* __restrict__ Wr,
                   const float* __restrict__ bias,
                   const float* __restrict__ Wc,
                   float* __restrict__ out,
                   int n_nodes, int n_tiles) {
    __shared__ float lWl[64 * 64];
    __shared__ float lWr[64 * 64];
    __shared__ float lb[64];
    __shared__ float lwc[64];

    const int tid = threadIdx.x;
    {
        const float4* gl = (const float4*)Wl;
        const float4* gr = (const float4*)Wr;
        float4* sl = (float4*)lWl;
        float4* sr = (float4*)lWr;
        for (int i = tid; i < 1024; i += 256) { sl[i] = gl[i]; sr[i] = gr[i]; }
        if (tid < 64)  lb[tid]  = bias[tid];
        if (tid >= 64 && tid < 128) lwc[tid - 64] = Wc[tid - 64];
    }
    __syncthreads();

    const int lane = tid & 31;
    const int wave = tid >> 5;
    const int n0   = (wave & 3) * 16;
    const int mg   = wave >> 2;
    const int col  = lane & 15;
    const int kp   = (lane >> 4) * 2;

    float bl0[16], bl1[16], br0[16], br1[16];
#pragma unroll
    for (int ks = 0; ks < 16; ++ks) {
        const int k = 4 * ks + kp;
        bl0[ks] = lWl[(k + 0) * 64 + n0 + col];
        bl1[ks] = lWl[(k + 1) * 64 + n0 + col];
        br0[ks] = lWr[(k + 0) * 64 + n0 + col];
        br1[ks] = lWr[(k + 1) * 64 + n0 + col];
    }
    const float bv  = lb[n0 + col];
    const float wcv = lwc[n0 + col];

    for (int tile = blockIdx.x * 2 + mg; tile < n_tiles; tile += gridDim.x * 2) {
        const int m0 = tile * 16;
        int rowi = m0 + col;
        if (rowi >= n_nodes) rowi = n_nodes - 1;
        const float inv = 1.0f / fmaxf(deg[rowi], 1.0f);
        const float* ap = agg   + (size_t)rowi * 64 + kp;
        const float* xp = xroot + (size_t)rowi * 64 + kp;

        v8f acc = {};
#pragma unroll
        for (int ks = 0; ks < 16; ++ks) {
            v2f a = *(const v2f*)(ap + 4 * ks);
            a *= inv;
            v2f bl = { bl0[ks], bl1[ks] };
            acc = wmma4(a, bl, acc);
            v2f ax = *(const v2f*)(xp + 4 * ks);
            v2f br = { br0[ks], br1[ks] };
            acc = wmma4(ax, br, acc);
        }

        const int mhi = (lane >> 4) * 8;
#pragma unroll
        for (int r = 0; r < 8; ++r) {
            float v = fmaxf(acc[r] + bv, 0.0f) * wcv;
            // reduce the 16-lane N slice (xor masks stay within each half-wave)
            v += __shfl_xor(v, 1);
            v += __shfl_xor(v, 2);
            v += __shfl_xor(v, 4);
            v += __shfl_xor(v, 8);
            const int row = m0 + mhi + r;
            if ((lane & 15) == 0 && row < n_nodes) atomicAdd(out + row, v);
        }
    }
}

// ---------------------------------------------------------------------------
extern "C" void kernel_launch(void* const* d_in, const int* in_sizes, int n_in,
                              void* d_out, int out_size, void* d_ws, size_t ws_size,
                              hipStream_t stream) {
    const float* x   = (const float*)d_in[0];
    const int*   ei  = (const int*)d_in[1];
    const float* W1l = (const float*)d_in[2];
    const float* W1r = (const float*)d_in[3];
    const float* b1  = (const float*)d_in[4];
    const float* W2l = (const float*)d_in[5];
    const float* W2r = (const float*)d_in[6];
    const float* b2  = (const float*)d_in[7];
    const float* Wc  = (const float*)d_in[8];
    const float* bc  = (const float*)d_in[9];
    float* out = (float*)d_out;

    const int n_nodes = in_sizes[0] / 64;
    const int n_edges = in_sizes[1] / 2;
    const int* src = ei;             // edge_index[0]
    const int* dst = ei + n_edges;   // edge_index[1]

    // workspace: deg [N] | agg [N*64] | h1 [N*64]   (~52 MB for N=100k)
    const int n_pad = (n_nodes + 3) & ~3;           // keep float4 alignment
    float* deg = (float*)d_ws;
    float* agg = deg + n_pad;
    float* h1  = agg + (size_t)n_nodes * 64;

    hipMemsetAsync(deg, 0, sizeof(float) * (size_t)n_pad, stream);
    hipMemsetAsync(agg, 0, sizeof(float) * (size_t)n_nodes * 64, stream);

    const int sc_blocks = (n_edges * 16 + 255) / 256;
    const int n_tiles = (n_nodes + 15) / 16;
    const int pairs = (n_tiles + 1) / 2;
    const int gblocks = pairs < 1280 ? pairs : 1280;

    // Layer 1: scatter x -> agg (+deg), then WMMA GEMM -> h1
    scatter_accum<<<sc_blocks, 256, 0, stream>>>(x, src, dst, agg, deg, n_edges);
    sage_gemm<<<gblocks, 256, 0, stream>>>(x, agg, deg, W1l, W1r, b1, h1,
                                           n_nodes, n_tiles);

    // Layer 2: re-zero agg, scatter h1 -> agg, fused GEMM+classifier -> out
    hipMemsetAsync(agg, 0, sizeof(float) * (size_t)n_nodes * 64, stream);
    scatter_accum<<<sc_blocks, 256, 0, stream>>>(h1, src, dst, agg, nullptr, n_edges);
    init_out<<<(n_nodes + 255) / 256, 256, 0, stream>>>(out, bc, n_nodes);
    sage_gemm_cls<<<gblocks, 256, 0, stream>>>(h1, agg, deg, W2l, W2r, b2, Wc, out,
                                               n_nodes, n_tiles);
}